// HebbianTrace_22445499089606
// MI455X (gfx1250) — compile-verified
//
#include <hip/hip_runtime.h>
#include <math.h>

typedef __attribute__((ext_vector_type(2))) float v2f;
typedef __attribute__((ext_vector_type(8))) float v8f;

#define B_  2
#define H_  16
#define T_  2048
#define D_  64
#define R_  8
#define BH_ (B_*H_)
#define NT_ (T_/16)   // 128 16-wide tiles per T dimension

// ---------------------------------------------------------------------------
// Kernel 1: low-rank projections  q_r = q @ Wq^T,  k_r = k @ Wk^T
// One thread per output element (row = bh*T+t, col = r). Threads sharing a row
// re-read the same 64-float q/k row -> served from cache; Wq/Wk are 2 KB.
// ---------------------------------------------------------------------------
__global__ void __launch_bounds__(256)
hebbian_proj_kernel(const float* __restrict__ q, const float* __restrict__ k,
                    const float* __restrict__ Wq, const float* __restrict__ Wk,
                    float* __restrict__ qr, float* __restrict__ kr) {
    int tid = blockIdx.x * blockDim.x + threadIdx.x;
    const int n = BH_ * T_ * R_;
    bool isK = (tid >= n);
    int idx = isK ? (tid - n) : tid;
    if (idx >= n) return;
    int r   = idx & (R_ - 1);
    int row = idx >> 3;                       // bh*T + t
    const float* src = (isK ? k : q) + (size_t)row * D_;
    const float* w   = (isK ? Wk : Wq) + (size_t)r * D_;
    float acc = 0.0f;
#pragma unroll
    for (int d = 0; d < D_; ++d) acc = fmaf(src[d], w[d], acc);
    (isK ? kr : qr)[idx] = acc;
}

// ---------------------------------------------------------------------------
// Kernel 2: causal EMA over time, one thread per (b,h,r) channel (256 total).
//   ema_t = decay * ema_{t-1} + k_r[t]
// ---------------------------------------------------------------------------
__global__ void __launch_bounds__(256)
hebbian_ema_kernel(const float* __restrict__ kr, float* __restrict__ ema,
                   const float* __restrict__ log_decay) {
    int c = blockIdx.x * blockDim.x + threadIdx.x;
    if (c >= BH_ * R_) return;
    int bh = c >> 3;
    int r  = c & 7;
    float ld    = log_decay[0];
    float decay = 1.0f / (1.0f + expf(-ld));   // sigmoid
    float e = 0.0f;
    size_t base = (size_t)bh * T_ * R_ + r;
    for (int t = 0; t < T_; ++t) {
        size_t i = base + (size_t)t * R_;
        e = fmaf(decay, e, kr[i]);
        ema[i] = e;
    }
}

// ---------------------------------------------------------------------------
// Kernel 3: trace_bias = (q_r @ ema^T) * scale with causal mask, via WMMA.
// One wave per 16x16 output tile; K=8 done as two V_WMMA_F32_16X16X4_F32.
//
// Fragment layouts (CDNA5 ISA, 32-bit operands, wave32):
//   A 16x4 : lane L holds (M = L&15, K = 2*(L>>4) + v) in VGPR v (v=0,1)
//   B 4x16 : lane L holds (K = 2*(L>>4) + v, N = L&15) in VGPR v
//   C 16x16: lane L, VGPR v holds (M = v + 8*(L>>4), N = L&15)
// Both A and B fragments are contiguous float2 loads from row-major (T,8).
// ---------------------------------------------------------------------------
__global__ void __launch_bounds__(256)
hebbian_trace_kernel(const float* __restrict__ qr, const float* __restrict__ ema,
                     const float* __restrict__ scale_p, float* __restrict__ out) {
    const int lane = threadIdx.x & 31;
    const int wave = threadIdx.x >> 5;
    const int half = lane >> 4;     // 0: lanes 0-15, 1: lanes 16-31
    const int lr   = lane & 15;     // N (column within tile)

    const int j_tile = blockIdx.x * 8 + wave;   // 0..127
    const int i_tile = blockIdx.y;              // 0..127
    const int bh     = blockIdx.z;              // 0..31

    float* outp = out + ((size_t)bh * T_ + (size_t)i_tile * 16) * T_
                      + (size_t)j_tile * 16;

    if (j_tile > i_tile) {
        // Strictly upper-triangular tile: all zeros (wave-uniform branch).
#pragma unroll
        for (int v = 0; v < 8; ++v)
            outp[(size_t)(v + 8 * half) * T_ + lr] = 0.0f;
        return;
    }

    const float* qrow = qr  + ((size_t)bh * T_ + (size_t)i_tile * 16 + lr) * R_ + 2 * half;
    const float* ecol = ema + ((size_t)bh * T_ + (size_t)j_tile * 16 + lr) * R_ + 2 * half;

    v2f a0 = *(const v2f*)(qrow);       // K = 0..3 slice
    v2f a1 = *(const v2f*)(qrow + 4);   // K = 4..7 slice
    v2f b0 = *(const v2f*)(ecol);
    v2f b1 = *(const v2f*)(ecol + 4);

    v8f c = {0.0f, 0.0f, 0.0f, 0.0f, 0.0f, 0.0f, 0.0f, 0.0f};
    c = __builtin_amdgcn_wmma_f32_16x16x4_f32(false, a0, false, b0,
                                              (short)0, c, false, false);
    c = __builtin_amdgcn_wmma_f32_16x16x4_f32(false, a1, false, b1,
                                              (short)0, c, false, false);

    const float s = scale_p[0];
    if (i_tile == j_tile) {
        // Diagonal tile: zero where column > row within the tile.
#pragma unroll
        for (int v = 0; v < 8; ++v) {
            int rowm = v + 8 * half;
            float val = (lr > rowm) ? 0.0f : c[v] * s;
            outp[(size_t)rowm * T_ + lr] = val;
        }
    } else {
#pragma unroll
        for (int v = 0; v < 8; ++v)
            outp[(size_t)(v + 8 * half) * T_ + lr] = c[v] * s;
    }
}

// ---------------------------------------------------------------------------
// Host-side launch
// ---------------------------------------------------------------------------
extern "C" void kernel_launch(void* const* d_in, const int* in_sizes, int n_in,
                              void* d_out, int out_size, void* d_ws, size_t ws_size,
                              hipStream_t stream) {
    const float* q         = (const float*)d_in[0];  // (B,H,T,D)
    const float* k         = (const float*)d_in[1];  // (B,H,T,D)
    const float* Wq        = (const float*)d_in[2];  // (R,D)
    const float* Wk        = (const float*)d_in[3];  // (R,D)
    const float* log_decay = (const float*)d_in[4];  // scalar
    const float* scale     = (const float*)d_in[5];  // scalar
    float* out = (float*)d_out;                      // (B,H,T,T)

    const int nProj = BH_ * T_ * R_;                 // 524288 elements each
    float* qr  = (float*)d_ws;
    float* kr  = qr + nProj;
    float* ema = kr + nProj;                         // total 6 MB of workspace

    // 1) projections (both q_r and k_r in one launch)
    int n2 = 2 * nProj;
    hebbian_proj_kernel<<<(n2 + 255) / 256, 256, 0, stream>>>(q, k, Wq, Wk, qr, kr);

    // 2) causal EMA scan (256 channels)
    hebbian_ema_kernel<<<1, 256, 0, stream>>>(kr, ema, log_decay);

    // 3) WMMA tile GEMM + mask + scale (one wave per 16x16 tile)
    dim3 grid(NT_ / 8, NT_, BH_);
    hebbian_trace_kernel<<<grid, 256, 0, stream>>>(qr, ema, scale, out);
}